// SoftDTWDivergence_72550587564583
// MI455X (gfx1250) — compile-verified
//
#include <hip/hip_runtime.h>
#include <hip/hip_bf16.h>

typedef float v2f __attribute__((ext_vector_type(2)));
typedef float v4f __attribute__((ext_vector_type(4)));
typedef float v8f __attribute__((ext_vector_type(8)));

#define BIGV 1e10f
#define TLEN 256
#define KDIM 64
#define NBATCH 512

// One workgroup per (pair, batch) problem.
//  Phase 1: ha[i]=0.5||a_i||^2, hb[j]=0.5||b_j||^2 (per-thread, float4 loads).
//  Phase 2: dot(a_i, b_j) for the whole 256x256 matrix via V_WMMA_F32_16X16X4_F32.
//           Each wave owns two 16-row strips; A fragments for a strip are loaded
//           once into registers (16 x v2f) and reused across all 16 column tiles.
//           tj/strip loops are pinned to no-unroll so only one accumulator and one
//           tile's B fragments are live -> no spills, pipelined loadcnt waits.
//  Phase 3: soft-DTW anti-diagonal DP, 256 threads = 256 rows, one barrier/step.
__global__ __launch_bounds__(256) void sdtw_pair_kernel(
    const float* __restrict__ X, const float* __restrict__ Y,
    float* __restrict__ partial)
{
    __shared__ float Cdot[TLEN * TLEN];   // 256 KB: dot products, LDS-resident
    __shared__ float ha[TLEN];
    __shared__ float hb[TLEN];
    __shared__ float dbuf[2][TLEN];       // DP wavefront exchange (parity buffered)

    const int wg = blockIdx.x;
    const int p  = wg / NBATCH;           // 0: s(X,Y)  1: s(X,X)  2: s(Y,Y)
    const int b  = wg % NBATCH;

    const float* Abase = (p == 2 ? Y : X) + (size_t)b * TLEN * KDIM;
    const float* Bbase = (p == 1 ? X : Y) + (size_t)b * TLEN * KDIM;

    const int tid  = threadIdx.x;
    const int lane = tid & 31;
    const int wave = tid >> 5;

    // ---- Phase 1: half squared norms (one row per thread, float4 loads) ----
    {
        const v4f* ar = (const v4f*)(Abase + tid * KDIM);
        const v4f* br = (const v4f*)(Bbase + tid * KDIM);
        float sa = 0.0f, sb = 0.0f;
        #pragma unroll
        for (int k = 0; k < KDIM / 4; ++k) {
            v4f a = ar[k], c = br[k];
            sa += a.x * a.x + a.y * a.y + a.z * a.z + a.w * a.w;
            sb += c.x * c.x + c.y * c.y + c.z * c.z + c.w * c.w;
        }
        ha[tid] = 0.5f * sa;
        hb[tid] = 0.5f * sb;
    }

    // ---- Phase 2: Cdot = A * B^T with fp32 WMMA, A register-blocked per strip ----
    // A-matrix 16x4 f32 layout: lane L holds A[M=L&15][K = 2*(L>>4) + vgpr]
    // B-matrix 4x16 f32 layout: lane L holds B[K = 2*(L>>4) + vgpr][N=L&15] = Brow[L&15][k]
    // -> both operands are the same float2 load from row-major (row, k) storage.
    {
        const int m  = lane & 15;
        const int kh = (lane >> 4) * 2;
        #pragma unroll 1
        for (int strip = 0; strip < 2; ++strip) {
            const int ti = wave + strip * 8;          // 16 strips over 8 waves x 2
            const float* arow = Abase + (ti * 16 + m) * KDIM + kh;
            v2f afrag[KDIM / 4];                       // 32 VGPRs, live across tj loop
            #pragma unroll
            for (int kb = 0; kb < KDIM / 4; ++kb)
                afrag[kb] = *(const v2f*)(arow + kb * 4);

            #pragma unroll 1
            for (int tj = 0; tj < 16; ++tj) {
                const float* brow = Bbase + (tj * 16 + m) * KDIM + kh;
                v8f acc = {};
                #pragma unroll
                for (int kb = 0; kb < KDIM / 4; ++kb) {   // 16 x v_wmma_f32_16x16x4_f32
                    v2f bv = *(const v2f*)(brow + kb * 4);
                    acc = __builtin_amdgcn_wmma_f32_16x16x4_f32(
                        false, afrag[kb], false, bv, (short)0, acc, false, false);
                }
                // C/D layout: lanes 0-15 -> rows v, lanes 16-31 -> rows v+8; col = lane&15
                const int row0 = ti * 16 + (lane >> 4) * 8;
                const int col  = tj * 16 + m;
                #pragma unroll
                for (int v = 0; v < 8; ++v)
                    Cdot[(row0 + v) * TLEN + col] = acc[v];
            }
        }
    }
    __syncthreads();

    // ---- Phase 3: soft-DTW DP, anti-diagonal wavefront ----
    // Thread i owns row i. At step s it computes V[i][s-i].
    //   left = own previous value (register)
    //   up   = dbuf[(s-1)&1][i-1]   (neighbor's step s-1 result)
    //   diag = neighbor's step s-2 result == previous step's `up` (register)
    // Cdot read on step s hits LDS bank (s-i) mod 64 across threads -> conflict-free.
    const int i = tid;
    float vleft = BIGV;
    float vdiag = (i == 0) ? 0.0f : BIGV;   // V[-1][-1] = 0, V[i-1][-1] = inf
    float myv = 0.0f;
    for (int s = 0; s < 2 * TLEN - 1; ++s) {
        const int j = s - i;
        if (j >= 0 && j < TLEN) {
            const float up = (i == 0) ? BIGV : dbuf[(s + 1) & 1][i - 1];
            const float c  = ha[i] + hb[j] - Cdot[i * TLEN + j];
            // softmin_gamma (gamma=1), max-shifted for stability
            const float m0 = fminf(vleft, fminf(vdiag, up));
            const float ss = __expf(m0 - vleft) + __expf(m0 - vdiag) + __expf(m0 - up);
            const float v  = c + m0 - __logf(ss);
            dbuf[s & 1][i] = v;
            vdiag = up;
            vleft = v;
            myv   = v;
        }
        __syncthreads();
    }

    if (i == TLEN - 1) {
        // mean over batch of s_xy - 0.5 s_xx - 0.5 s_yy
        const float w = (p == 0) ? (1.0f / (float)NBATCH) : (-0.5f / (float)NBATCH);
        partial[wg] = w * myv;
    }
}

// Deterministic tree reduction of the 1536 partials -> scalar
__global__ __launch_bounds__(256) void sdtw_reduce_kernel(
    const float* __restrict__ partial, float* __restrict__ out, int n)
{
    __shared__ float s[256];
    const int tid = threadIdx.x;
    float acc = 0.0f;
    for (int t = tid; t < n; t += 256) acc += partial[t];
    s[tid] = acc;
    __syncthreads();
    for (int off = 128; off > 0; off >>= 1) {
        if (tid < off) s[tid] += s[tid + off];
        __syncthreads();
    }
    if (tid == 0) out[0] = s[0];
}

extern "C" void kernel_launch(void* const* d_in, const int* in_sizes, int n_in,
                              void* d_out, int out_size, void* d_ws, size_t ws_size,
                              hipStream_t stream) {
    const float* X = (const float*)d_in[0];   // (512,256,64) f32
    const float* Y = (const float*)d_in[1];   // (512,256,64) f32
    float* out     = (float*)d_out;           // scalar
    float* partial = (float*)d_ws;            // 3*512 floats = 6 KB scratch

    const int nprob = 3 * NBATCH;             // 1536 independent DP problems
    sdtw_pair_kernel<<<nprob, 256, 0, stream>>>(X, Y, partial);
    sdtw_reduce_kernel<<<1, 256, 0, stream>>>(partial, out, nprob);
}